// ChannelAttetionPropagation1D_40922448396289
// MI455X (gfx1250) — compile-verified
//
#include <hip/hip_runtime.h>
#include <math.h>

typedef float v2f __attribute__((ext_vector_type(2)));
typedef float v8f __attribute__((ext_vector_type(8)));

#define NB 4
#define TT 8
#define NROWS 16384
#define CC 64
#define KTOT (TT * NROWS)                                   // 131072 reduction rows per batch
#define BLK_PER_B 128
#define WAVES_PER_BLK 8
#define ROWS_PER_WAVE (KTOT / (BLK_PER_B * WAVES_PER_BLK))  // 128
#define KSTEPS (ROWS_PER_WAVE / 4)                          // 32 steps of K=4

// ---------------------------------------------------------------------------
// Kernel 0: zero the 4 x 64 x 64 fp32 accumulator in workspace
// ---------------------------------------------------------------------------
__global__ __launch_bounds__(256) void zero_ws_kernel(float* __restrict__ p, int nelems) {
  int i = blockIdx.x * 256 + threadIdx.x;
  if (i < nelems) p[i] = 0.0f;
}

// ---------------------------------------------------------------------------
// Kernel 1: kv_mul[b,c,d] = sum_{k in T*n} key_mem[b,k,c] * val_mem[b,k,d]
// Split-K across 128 blocks x 8 waves per batch; each wave accumulates the
// full 64x64 tile with V_WMMA_F32_16X16X4_F32 (4x4 tiles of 16x16).
// Wave partials -> ds_add_f32 into LDS -> one global_atomic_add_f32 pass.
// ---------------------------------------------------------------------------
__global__ __launch_bounds__(256) void kv_reduce_kernel(
    const float* __restrict__ key_mem, const float* __restrict__ val_mem,
    float* __restrict__ kv_acc) {
  __shared__ float lds[64 * 64];

  const int tid  = threadIdx.x;
  const int lane = tid & 31;
  const int wid  = tid >> 5;
  const int b    = blockIdx.x >> 7;   // 128 blocks per batch
  const int blk  = blockIdx.x & 127;
  const int lh   = lane >> 4;         // lane-half (0/1)
  const int ll   = lane & 15;

  for (int i = tid; i < 64 * 64; i += 256) lds[i] = 0.0f;
  __syncthreads();

  const float* kbase = key_mem + (size_t)b * KTOT * CC;
  const float* vbase = val_mem + (size_t)b * KTOT * CC;
  size_t kr = (size_t)(blk * WAVES_PER_BLK + wid) * ROWS_PER_WAVE;

  v8f acc[4][4];
#pragma unroll
  for (int ct = 0; ct < 4; ++ct)
#pragma unroll
    for (int dt = 0; dt < 4; ++dt) acc[ct][dt] = (v8f)0.0f;

  // fragment register j at lane-half h holds K = 2h + j (ISA 16x4 f32 layout)
  const int foff = (2 * lh) * CC + ll;  // + tile*16 ; second reg is +CC

#pragma unroll 2
  for (int s = 0; s < KSTEPS; ++s, kr += 4) {
    const float* krow = kbase + kr * CC;
    const float* vrow = vbase + kr * CC;
    if (s + 4 < KSTEPS) {  // stream-ahead prefetch (global_prefetch_b8)
      __builtin_prefetch(krow + 16 * CC + foff, 0, 1);
      __builtin_prefetch(vrow + 16 * CC + foff, 0, 1);
    }
    v2f af[4], bf[4];
#pragma unroll
    for (int ct = 0; ct < 4; ++ct) {
      af[ct].x = krow[foff + ct * 16];            // A[M=ct*16+ll, K=2h]
      af[ct].y = krow[foff + ct * 16 + CC];       // A[M, K=2h+1]
    }
#pragma unroll
    for (int dt = 0; dt < 4; ++dt) {
      bf[dt].x = vrow[foff + dt * 16];            // B[K=2h,   N=dt*16+ll]
      bf[dt].y = vrow[foff + dt * 16 + CC];       // B[K=2h+1, N]
    }
#pragma unroll
    for (int ct = 0; ct < 4; ++ct)
#pragma unroll
      for (int dt = 0; dt < 4; ++dt)
        acc[ct][dt] = __builtin_amdgcn_wmma_f32_16x16x4_f32(
            false, af[ct], false, bf[dt], (short)0, acc[ct][dt], false, false);
  }

  // wave partials -> LDS (ds_add_f32)
#pragma unroll
  for (int ct = 0; ct < 4; ++ct)
#pragma unroll
    for (int dt = 0; dt < 4; ++dt)
#pragma unroll
      for (int r = 0; r < 8; ++r) {
        int M = ct * 16 + r + 8 * lh;   // C/D layout: reg r, lane-half h -> M = r + 8h
        int Nn = dt * 16 + ll;
        __hip_atomic_fetch_add(&lds[M * 64 + Nn], acc[ct][dt][r],
                               __ATOMIC_RELAXED, __HIP_MEMORY_SCOPE_WORKGROUP);
      }
  __syncthreads();

  float* outp = kv_acc + b * 64 * 64;
  for (int i = tid; i < 64 * 64; i += 256)
    __hip_atomic_fetch_add(&outp[i], lds[i],
                           __ATOMIC_RELAXED, __HIP_MEMORY_SCOPE_AGENT);
}

// ---------------------------------------------------------------------------
// Kernel 2 (fused): softmax over C of kv_acc[b] in LDS, then
// out[b,i,d] = alpha * sum_c key_cur[b,i,c] * P[c,d] + val_cur[b,i,d]
// Each wave: one 16-row tile, 4 d-tiles, K=64 via 16 WMMA f32 steps.
// ---------------------------------------------------------------------------
#define PPAD 68  // 64 + 4 pad: kills LDS bank conflicts, keeps 16B alignment

__global__ __launch_bounds__(256) void attn_apply_kernel(
    const float* __restrict__ key_cur, const float* __restrict__ val_cur,
    const float* __restrict__ kv_acc, const float* __restrict__ alpha_p,
    float* __restrict__ out) {
  __shared__ float P[64 * PPAD];               // softmaxed kv tile
  __shared__ float Kt[WAVES_PER_BLK][16 * PPAD];  // per-wave key_cur tile

  const int tid  = threadIdx.x;
  const int lane = tid & 31;
  const int wid  = tid >> 5;
  const int b    = blockIdx.x >> 7;   // 128 blocks per batch
  const int blk  = blockIdx.x & 127;
  const int lh   = lane >> 4;
  const int ll   = lane & 15;
  const float alpha = alpha_p[0];

  // load raw 64x64 accumulator into padded LDS
  const float* kvb = kv_acc + b * 64 * 64;
  for (int i = tid; i < 64 * 64; i += 256) {
    int c = i >> 6, d = i & 63;
    P[c * PPAD + d] = kvb[i];
  }
  __syncthreads();

  // softmax over the C axis (one thread per column d)
  if (tid < 64) {
    const int d = tid;
    float mx = -INFINITY;
    for (int c = 0; c < 64; ++c) mx = fmaxf(mx, P[c * PPAD + d]);
    float s = 0.0f;
    for (int c = 0; c < 64; ++c) {
      float e = __expf(P[c * PPAD + d] - mx);
      P[c * PPAD + d] = e;
      s += e;
    }
    float inv = 1.0f / s;
    for (int c = 0; c < 64; ++c) P[c * PPAD + d] *= inv;
  }

  // stage this wave's 16x64 key_cur tile into LDS (coalesced float4 loads)
  const int i0 = (blk * WAVES_PER_BLK + wid) * 16;  // row within batch
  const float* krow = key_cur + ((size_t)b * NROWS + i0) * CC;
  float* kt = Kt[wid];
#pragma unroll
  for (int it = 0; it < 8; ++it) {
    int idx = it * 32 + lane;          // 256 float4 elements total
    int r = idx >> 4, c4 = idx & 15;
    float4 v = ((const float4*)(krow + r * CC))[c4];
    *(float4*)&kt[r * PPAD + c4 * 4] = v;
  }
  __syncthreads();

  v8f acc[4];
#pragma unroll
  for (int dt = 0; dt < 4; ++dt) acc[dt] = (v8f)0.0f;

#pragma unroll
  for (int k = 0; k < CC; k += 4) {
    // A[M=ll, K=k+2h+j] from staged key tile (ds_load, conflict-free)
    v2f a = *(const v2f*)&kt[ll * PPAD + k + 2 * lh];
#pragma unroll
    for (int dt = 0; dt < 4; ++dt) {
      v2f bb;
      bb.x = P[(k + 2 * lh) * PPAD + dt * 16 + ll];
      bb.y = P[(k + 2 * lh + 1) * PPAD + dt * 16 + ll];
      acc[dt] = __builtin_amdgcn_wmma_f32_16x16x4_f32(
          false, a, false, bb, (short)0, acc[dt], false, false);
    }
  }

  // epilogue: out = alpha * acc + val_cur
  const float* vrow = val_cur + ((size_t)b * NROWS + i0) * CC;
  float* orow = out + ((size_t)b * NROWS + i0) * CC;
#pragma unroll
  for (int dt = 0; dt < 4; ++dt)
#pragma unroll
    for (int r = 0; r < 8; ++r) {
      int row = r + 8 * lh;
      int col = dt * 16 + ll;
      orow[row * CC + col] = alpha * acc[dt][r] + vrow[row * CC + col];
    }
}

// ---------------------------------------------------------------------------
extern "C" void kernel_launch(void* const* d_in, const int* in_sizes, int n_in,
                              void* d_out, int out_size, void* d_ws, size_t ws_size,
                              hipStream_t stream) {
  const float* key_mem = (const float*)d_in[0];  // (4, 8, 16384, 64)
  const float* val_mem = (const float*)d_in[1];  // (4, 8, 16384, 64)
  const float* key_cur = (const float*)d_in[2];  // (4, 16384, 64)
  const float* val_cur = (const float*)d_in[3];  // (4, 16384, 64)
  const float* alpha   = (const float*)d_in[4];  // (1,)
  float* outp   = (float*)d_out;                 // (4, 16384, 64)
  float* kv_acc = (float*)d_ws;                  // 4*64*64 fp32 accumulator

  const int acc_elems = NB * 64 * 64;  // 16384
  zero_ws_kernel<<<(acc_elems + 255) / 256, 256, 0, stream>>>(kv_acc, acc_elems);
  kv_reduce_kernel<<<NB * BLK_PER_B, 256, 0, stream>>>(key_mem, val_mem, kv_acc);
  attn_apply_kernel<<<NB * BLK_PER_B, 256, 0, stream>>>(key_cur, val_cur, kv_acc, alpha, outp);
}